// AutoEncoderGRU_8985071583719
// MI455X (gfx1250) — compile-verified
//
#include <hip/hip_runtime.h>

// CDNA5 / gfx1250 GRU scan.
// Formulation: per wave, 16 sequences. Each step:
//   D(16x16,f32) = A(16x4) @ B(4x16) + C
//     A rows 0..8 : [ W_hh (K=0..2) | W_ih rows 0..5 in K=3 ]
//     B cols 0..15: [ h0 ; h1 ; h2 ; x_t ]   (K rows, col n = sequence n)
//     C rows      : b_hh (+ b_ih for rows 0..5)
//   -> D rows 0..5 = pre-activations of r,z gates (x fused), rows 6..8 = hn.
// Lanes 0..15 each own one sequence: gates + update are plain VALU.
// Cross-half traffic: 2x ds_bpermute per step (gate row 8, h2 for next B).

typedef __attribute__((ext_vector_type(2))) float v2f;
typedef __attribute__((ext_vector_type(8))) float v8f;

__device__ __forceinline__ float fast_sigmoid(float x) {
  // 1 / (1 + 2^(-x*log2 e))
  float e = __builtin_amdgcn_exp2f(x * -1.44269504088896340736f);
  return __builtin_amdgcn_rcpf(1.0f + e);
}
__device__ __forceinline__ float fast_tanh(float x) {
  // 2 / (1 + 2^(-2x*log2 e)) - 1
  float e = __builtin_amdgcn_exp2f(x * -2.88539008177792681472f);
  float r = __builtin_amdgcn_rcpf(1.0f + e);
  return __builtin_fmaf(2.0f, r, -1.0f);
}

__global__ __launch_bounds__(256) void gru_scan_wmma(
    const float* __restrict__ x,      // [B,T]
    const int*   __restrict__ slen,   // [B]
    const float* __restrict__ h0p,    // [B,3]
    const float* __restrict__ W_ih,   // [9,1]
    const float* __restrict__ W_hh,   // [9,3]
    const float* __restrict__ b_ih,   // [9]
    const float* __restrict__ b_hh,   // [9]
    float* __restrict__ out,          // [B,3]
    int B, int T)
{
  const int lane  = threadIdx.x & 31;
  const int wave  = (blockIdx.x * blockDim.x + threadIdx.x) >> 5;
  const int m     = lane & 15;          // matrix row (A) / column (B,D)
  const bool hi   = lane >= 16;
  const int batch = wave * 16 + m;
  const bool valid = batch < B;

  // ---- n-gate x-projection weights (uniform scalars) ----
  const float wi6 = W_ih[6], wi7 = W_ih[7], wi8 = W_ih[8];
  const float bi6 = b_ih[6], bi7 = b_ih[7], bi8 = b_ih[8];

  // ---- A operand: 16x4 f32, VGPR0 = K(0|2), VGPR1 = K(1|3) per lane half ----
  float a0 = 0.0f, a1 = 0.0f;
  if (m < 9) {
    if (!hi) { a0 = W_hh[m * 3 + 0]; a1 = W_hh[m * 3 + 1]; }
    else     { a0 = W_hh[m * 3 + 2]; a1 = (m < 6) ? W_ih[m] : 0.0f; }
  }
  v2f A; A.x = a0; A.y = a1;

  // ---- C operand: row r -> b_hh[r] (+ b_ih[r] for fused r/z rows) ----
  v8f Cb;
#pragma unroll
  for (int j = 0; j < 8; ++j) {
    const int row = hi ? (j + 8) : j;
    float c = 0.0f;
    if (row < 9) { c = b_hh[row]; if (row < 6) c += b_ih[row]; }
    Cb[j] = c;
  }

  // ---- sequence lengths; wave-uniform loop bound via swizzle-max ----
  int len = valid ? slen[batch] : 0;
  int mx = len, o;
  o = __builtin_amdgcn_ds_swizzle(mx, 0x041F); mx = o > mx ? o : mx; // xor 1
  o = __builtin_amdgcn_ds_swizzle(mx, 0x081F); mx = o > mx ? o : mx; // xor 2
  o = __builtin_amdgcn_ds_swizzle(mx, 0x101F); mx = o > mx ? o : mx; // xor 4
  o = __builtin_amdgcn_ds_swizzle(mx, 0x201F); mx = o > mx ? o : mx; // xor 8
  o = __builtin_amdgcn_ds_swizzle(mx, 0x401F); mx = o > mx ? o : mx; // xor 16
  const int tmax = __builtin_amdgcn_readfirstlane(mx);

  // ---- hidden state (lanes 0..15 authoritative; h0 is zeros per reference) ----
  float h0v = 0.0f, h1v = 0.0f, h2v = 0.0f;
  if (valid) {
    h0v = h0p[batch * 3 + 0];
    h1v = h0p[batch * 3 + 1];
    h2v = h0p[batch * 3 + 2];
  }

  const int idx_lo = m << 2;                    // bpermute: broadcast lower half
  const int idx_hi = ((lane + 16) & 31) << 2;   // bpermute: fetch cross-half row 8

  const float* xb = x + (size_t)(valid ? batch : 0) * (size_t)T;
  const bool vec_ok = (T & 3) == 0;

  for (int tt = 0; tt < tmax; tt += 4) {
    float xs0 = 0.0f, xs1 = 0.0f, xs2 = 0.0f, xs3 = 0.0f;
    if (valid) {
      if (vec_ok && (tt + 4) <= T) {
        const float4 q = *(const float4*)(xb + tt);
        xs0 = q.x; xs1 = q.y; xs2 = q.z; xs3 = q.w;
        if (tt + 68 <= T) __builtin_prefetch(xb + tt + 64, 0, 0); // global_prefetch_b8
      } else {
        if (tt + 0 < T) xs0 = xb[tt + 0];
        if (tt + 1 < T) xs1 = xb[tt + 1];
        if (tt + 2 < T) xs2 = xb[tt + 2];
        if (tt + 3 < T) xs3 = xb[tt + 3];
      }
    }
    float xs[4] = {xs0, xs1, xs2, xs3};

#pragma unroll
    for (int u = 0; u < 4; ++u) {
      const int t = tt + u;
      const float xv = xs[u];

      // ---- build B operand: VGPR0 = (h0 | h2), VGPR1 = (h1 | x_t) ----
      const float h2all = __int_as_float(
          __builtin_amdgcn_ds_bpermute(idx_lo, __float_as_int(h2v)));
      v2f Bv;
      Bv.x = hi ? h2all : h0v;
      Bv.y = hi ? xv    : h1v;

      // ---- one WMMA = all 9 hidden projections + fused x/bias terms ----
      v8f D = __builtin_amdgcn_wmma_f32_16x16x4_f32(
          false, A, false, Bv, (short)0, Cb, false, false);

      // ---- gates (lanes 0..15 own rows 0..7; row 8 via cross-half permute) --
      const float r0 = fast_sigmoid(D[0]);
      const float r1 = fast_sigmoid(D[1]);
      const float r2 = fast_sigmoid(D[2]);
      const float z0 = fast_sigmoid(D[3]);
      const float z1 = fast_sigmoid(D[4]);
      const float z2 = fast_sigmoid(D[5]);
      const float hn2 = __int_as_float(
          __builtin_amdgcn_ds_bpermute(idx_hi, __float_as_int(D[0])));
      const float n0 = fast_tanh(__builtin_fmaf(r0, D[6], __builtin_fmaf(xv, wi6, bi6)));
      const float n1 = fast_tanh(__builtin_fmaf(r1, D[7], __builtin_fmaf(xv, wi7, bi7)));
      const float n2 = fast_tanh(__builtin_fmaf(r2, hn2, __builtin_fmaf(xv, wi8, bi8)));

      // h_new = (1-z)*n + z*h = n + z*(h-n); freeze past this sequence's length
      const float nh0 = __builtin_fmaf(z0, h0v - n0, n0);
      const float nh1 = __builtin_fmaf(z1, h1v - n1, n1);
      const float nh2 = __builtin_fmaf(z2, h2v - n2, n2);
      const bool act = t < len;
      h0v = act ? nh0 : h0v;
      h1v = act ? nh1 : h1v;
      h2v = act ? nh2 : h2v;
    }
  }

  if (!hi && valid) {
    out[batch * 3 + 0] = fast_sigmoid(h0v);
    out[batch * 3 + 1] = fast_sigmoid(h1v);
    out[batch * 3 + 2] = fast_sigmoid(h2v);
  }
}

extern "C" void kernel_launch(void* const* d_in, const int* in_sizes, int n_in,
                              void* d_out, int out_size, void* d_ws, size_t ws_size,
                              hipStream_t stream) {
  const float* x    = (const float*)d_in[0];
  const int*   sl   = (const int*)  d_in[1];
  const float* h0   = (const float*)d_in[2];
  const float* wih  = (const float*)d_in[3];
  const float* whh  = (const float*)d_in[4];
  const float* bih  = (const float*)d_in[5];
  const float* bhh  = (const float*)d_in[6];
  float* out = (float*)d_out;

  const int B = in_sizes[1];
  const int T = in_sizes[0] / (B > 0 ? B : 1);

  const int waves   = (B + 15) / 16;       // 16 sequences per wave32
  const int threads = waves * 32;
  const int block   = 256;                 // 8 waves per workgroup
  const int grid    = (threads + block - 1) / block;

  gru_scan_wmma<<<grid, block, 0, stream>>>(x, sl, h0, wih, whh, bih, bhh,
                                            out, B, T);
}